// Attention_67491116089467
// MI455X (gfx1250) — compile-verified
//
#include <hip/hip_runtime.h>
#include <math.h>

// Problem constants (match reference)
constexpr int kB = 64;
constexpr int kL = 1024;
constexpr int kH = 1024;

typedef __attribute__((ext_vector_type(2))) float v2f;
typedef __attribute__((ext_vector_type(4))) float v4f;
typedef __attribute__((ext_vector_type(8))) float v8f;

// ---------------------------------------------------------------------------
// Kernel 1: q = hidden @ W   (M=kB, N=kH, K=kH), fp32 WMMA 16x16x4.
// One wave (32 threads) computes one 16x16 tile of q.
//   A[m,k] = hidden[m*kH + k]      (hidden is [B,H], K dim = o)
//   B[k,n] = W[k*kH + n]           (W is [o,h] row-major)
// VGPR layouts per CDNA5 ISA 7.12.2:
//   A 16x4 : lanes 0-15 -> M=lane, K={0,1}; lanes 16-31 -> same M, K={2,3}
//   B 4x16 : N = lane%16; VGPR i holds K = i + 2*(lane/16)
//   C/D    : N = lane%16; VGPR i holds M = i + 8*(lane/16)
// ---------------------------------------------------------------------------
__global__ void __launch_bounds__(32)
qgemm_wmma_kernel(const float* __restrict__ hidden,
                  const float* __restrict__ W,
                  float* __restrict__ q) {
    const int lane  = threadIdx.x;       // 0..31, EXEC all ones (WMMA req.)
    const int lo    = lane & 15;
    const int half  = lane >> 4;         // 0 or 1
    const int nBase = blockIdx.x * 16;   // h tile
    const int mBase = blockIdx.y * 16;   // b tile

    v8f c = {0.f, 0.f, 0.f, 0.f, 0.f, 0.f, 0.f, 0.f};

    const float* arow = hidden + (size_t)(mBase + lo) * kH;  // A row m
    const float* bcol = W + (nBase + lo);                    // B column n

    #pragma unroll 4
    for (int k = 0; k < kH; k += 4) {
        // A fragment: K = k + 2*half + {0,1} -> one aligned 8B load
        v2f a = *(const v2f*)(arow + k + 2 * half);
        // B fragment: K = k + 2*half + {0,1}, stride kH between rows
        v2f b;
        b.x = bcol[(size_t)(k + 2 * half    ) * kH];
        b.y = bcol[(size_t)(k + 2 * half + 1) * kH];
        // 8-arg form: (neg_a, A, neg_b, B, c_mod, C, reuse_a, reuse_b)
        c = __builtin_amdgcn_wmma_f32_16x16x4_f32(
                false, a, false, b, (short)0, c, false, false);
    }

    // D layout: VGPR i -> row M = mBase + i + 8*half, col N = nBase + lo
    float* qt = q + (size_t)(mBase + 8 * half) * kH + nBase + lo;
    #pragma unroll
    for (int i = 0; i < 8; ++i)
        qt[(size_t)i * kH] = c[i];
}

// ---------------------------------------------------------------------------
// Kernel 2: cbias[b] = dot(hidden[b,:], bias).  One wave per b.
// ---------------------------------------------------------------------------
__global__ void __launch_bounds__(32)
bias_dot_kernel(const float* __restrict__ hidden,
                const float* __restrict__ bias,
                float* __restrict__ cbias) {
    const int b    = blockIdx.x;
    const int lane = threadIdx.x;
    const float* hrow = hidden + (size_t)b * kH;
    float acc = 0.f;
    #pragma unroll 4
    for (int h = lane; h < kH; h += 32)
        acc = fmaf(hrow[h], bias[h], acc);
    #pragma unroll
    for (int off = 16; off > 0; off >>= 1)
        acc += __shfl_xor(acc, off, 32);
    if (lane == 0) cbias[b] = acc;
}

// ---------------------------------------------------------------------------
// Kernel 3: energies[b,l] = enc[l,b,:] . q[b,:] + cbias[b]
// Bandwidth kernel: 256 MB of enc streamed once.  One wave per (l,b) pair,
// b128 coalesced loads; enc marked non-temporal (read-once) so the 256 KB q
// (re-read L times) stays resident in L2.
// ---------------------------------------------------------------------------
__global__ void __launch_bounds__(256)
energies_kernel(const float* __restrict__ enc,
                const float* __restrict__ q,
                const float* __restrict__ cbias,
                float* __restrict__ e) {
    const int wavesPerBlock = blockDim.x >> 5;                 // 8
    const int wave = blockIdx.x * wavesPerBlock + (threadIdx.x >> 5);
    const int lane = threadIdx.x & 31;
    const int l = wave / kB;
    const int b = wave % kB;

    const v4f* enc4 = (const v4f*)(enc + ((size_t)l * kB + b) * kH);
    const v4f* q4   = (const v4f*)(q + (size_t)b * kH);

    float acc = 0.f;
    #pragma unroll
    for (int j = 0; j < kH / (4 * 32); ++j) {                  // 8 iters
        v4f ev = __builtin_nontemporal_load(enc4 + lane + j * 32);
        v4f qv = q4[lane + j * 32];
        acc = fmaf(ev.x, qv.x, acc);
        acc = fmaf(ev.y, qv.y, acc);
        acc = fmaf(ev.z, qv.z, acc);
        acc = fmaf(ev.w, qv.w, acc);
    }
    #pragma unroll
    for (int off = 16; off > 0; off >>= 1)
        acc += __shfl_xor(acc, off, 32);
    if (lane == 0)
        e[(size_t)b * kL + l] = acc + cbias[b];
}

// ---------------------------------------------------------------------------
// Kernel 4: softmax over the BATCH axis (faithful to reference: axis=0).
// One thread per column l; walks b with stride kL (coalesced across threads).
// Output [B,1,L] flat == b*kL + l.
// ---------------------------------------------------------------------------
__global__ void __launch_bounds__(256)
softmax_batch_kernel(const float* __restrict__ e,
                     float* __restrict__ out) {
    const int l = blockIdx.x * blockDim.x + threadIdx.x;
    float m = -INFINITY;
    #pragma unroll 4
    for (int b = 0; b < kB; ++b)
        m = fmaxf(m, e[(size_t)b * kL + l]);
    float s = 0.f;
    #pragma unroll 4
    for (int b = 0; b < kB; ++b)
        s += __expf(e[(size_t)b * kL + l] - m);
    const float inv = 1.f / s;
    #pragma unroll 4
    for (int b = 0; b < kB; ++b)
        out[(size_t)b * kL + l] = __expf(e[(size_t)b * kL + l] - m) * inv;
}

// ---------------------------------------------------------------------------
extern "C" void kernel_launch(void* const* d_in, const int* in_sizes, int n_in,
                              void* d_out, int out_size, void* d_ws, size_t ws_size,
                              hipStream_t stream) {
    (void)in_sizes; (void)n_in; (void)out_size; (void)ws_size;

    const float* hidden = (const float*)d_in[0];   // [B,H]
    const float* enc    = (const float*)d_in[1];   // [L,B,H]
    const float* W      = (const float*)d_in[2];   // [H,H] ([out,in])
    const float* bias   = (const float*)d_in[3];   // [H]
    float* out = (float*)d_out;                    // [B,1,L]

    // Workspace layout (floats): q[kB*kH] | cbias[kB] | e[kB*kL]
    float* q     = (float*)d_ws;
    float* cbias = q + (size_t)kB * kH;
    float* e     = cbias + kB;

    // 1) q = hidden @ W via fp32 WMMA (one wave per 16x16 tile)
    qgemm_wmma_kernel<<<dim3(kH / 16, kB / 16), 32, 0, stream>>>(hidden, W, q);
    // 2) cbias = hidden @ bias
    bias_dot_kernel<<<kB, 32, 0, stream>>>(hidden, bias, cbias);
    // 3) energies (bandwidth-bound: 256 MB enc streamed once)
    energies_kernel<<<(kL * kB) / 8, 256, 0, stream>>>(enc, q, cbias, e);
    // 4) softmax over batch axis
    softmax_batch_kernel<<<kL / 256, 256, 0, stream>>>(e, out);
}